// DependencyMFVI_5652176961688
// MI455X (gfx1250) — compile-verified
//
#include <hip/hip_runtime.h>

typedef __attribute__((ext_vector_type(2))) float v2f;
typedef __attribute__((ext_vector_type(8))) float v8f;

#define NN   96     // n
#define BB   4      // batch
#define LDH  97     // padded LDS row stride (96+1) -> conflict-free column walks

// ---------------------------------------------------------------------------
// Row softmax over the last axis (h), one wave32 per row, 3 elements per lane.
// ---------------------------------------------------------------------------
__global__ __launch_bounds__(128) void mfvi_softmax(const float* __restrict__ in,
                                                    float* __restrict__ out,
                                                    int rows) {
    const int wave = threadIdx.x >> 5;
    const int lane = threadIdx.x & 31;
    const int row  = blockIdx.x * 4 + wave;
    if (row >= rows) return;

    const float* x = in + (size_t)row * NN;
    float v0 = x[lane];
    float v1 = x[lane + 32];
    float v2 = x[lane + 64];

    float mx = fmaxf(v0, fmaxf(v1, v2));
#pragma unroll
    for (int i = 16; i >= 1; i >>= 1) mx = fmaxf(mx, __shfl_xor(mx, i, 32));

    float e0 = __expf(v0 - mx);
    float e1 = __expf(v1 - mx);
    float e2 = __expf(v2 - mx);
    float sum = e0 + e1 + e2;
#pragma unroll
    for (int i = 16; i >= 1; i >>= 1) sum += __shfl_xor(sum, i, 32);

    float inv = 1.0f / sum;
    float* y = out + (size_t)row * NN;
    y[lane]      = e0 * inv;
    y[lane + 32] = e1 * inv;
    y[lane + 64] = e2 * inv;
}

// ---------------------------------------------------------------------------
// One workgroup per (b,d). Stage masked S[h][s] = s_sib_m[b,d,h,s] and
// Q[s][h] = qs[b,s,h] into LDS, then each of the 6 waves computes one
// diagonal 16x16 tile of S@Q via V_WMMA_F32_16X16X4_F32 (24 K=4 steps)
// and extracts the diagonal:  q_out[b,d,h] = s_arc[b,d,h] + (S@Q)[h,h].
// ---------------------------------------------------------------------------
__global__ __launch_bounds__(192) void mfvi_contract(
    const float* __restrict__ s_arc,        // [B,N,N]   (d,h)
    const float* __restrict__ s_sib,        // [B,N,N,N] (d,h,s)
    const unsigned char* __restrict__ mask, // [B,N] bool
    const float* __restrict__ qs,           // [B,N,N]   (s,h)
    float* __restrict__ q_out)              // [B,N,N]   (d,h)
{
    extern __shared__ float smem[];
    float* S    = smem;                 // [96][LDH]
    float* Q    = smem + NN * LDH;      // [96][LDH]
    float* mrow = Q + NN * LDH;         // [96]

    const int bd  = blockIdx.x;         // b*96 + d
    const int b   = bd / NN;
    const int d   = bd - b * NN;
    const int tid = threadIdx.x;        // 0..191 (6 full wave32s)

    if (tid < NN) {
        // root position (index 0) is always valid
        mrow[tid] = (tid == 0) ? 1.0f : (mask[b * NN + tid] ? 1.0f : 0.0f);
    }
    __syncthreads();
    const float md = mrow[d];

    const float* sib = s_sib + (size_t)bd * (NN * NN);  // [h][s] slice, contiguous
    const float* qsb = qs    + (size_t)b  * (NN * NN);  // [s][h] slice, contiguous

    // 2304 float4 vectors / 192 threads = 12 per thread (uniform -> EXEC stays full)
    for (int i = tid; i < (NN * NN) / 4; i += 192) {
        const int e  = i * 4;
        const int i0 = e / NN;          // row (h for S, s for Q)
        const int i1 = e - i0 * NN;     // col (s for S, h for Q); 96%4==0 -> same row

        float4 v = *(const float4*)(sib + e);
        const float mh = md * mrow[i0];
        const float f0 = mh * mrow[i1 + 0] * (((i1 + 0) != d && (i1 + 0) != i0) ? 1.0f : 0.0f);
        const float f1 = mh * mrow[i1 + 1] * (((i1 + 1) != d && (i1 + 1) != i0) ? 1.0f : 0.0f);
        const float f2 = mh * mrow[i1 + 2] * (((i1 + 2) != d && (i1 + 2) != i0) ? 1.0f : 0.0f);
        const float f3 = mh * mrow[i1 + 3] * (((i1 + 3) != d && (i1 + 3) != i0) ? 1.0f : 0.0f);
        S[i0 * LDH + i1 + 0] = v.x * f0;
        S[i0 * LDH + i1 + 1] = v.y * f1;
        S[i0 * LDH + i1 + 2] = v.z * f2;
        S[i0 * LDH + i1 + 3] = v.w * f3;

        float4 u = *(const float4*)(qsb + e);
        Q[i0 * LDH + i1 + 0] = u.x;
        Q[i0 * LDH + i1 + 1] = u.y;
        Q[i0 * LDH + i1 + 2] = u.z;
        Q[i0 * LDH + i1 + 3] = u.w;
    }
    __syncthreads();

    const int wave = tid >> 5;          // 0..5 -> h-tile
    const int lane = tid & 31;
    const int m    = lane & 15;         // A: row M; B/C/D: column N
    const int kh   = lane >> 4;         // K half-select for f32 A/B fragments
    const int h0   = wave * 16;

    // A fragment (16x4 f32): VGPR0 = K {0 | 2}, VGPR1 = K {1 | 3} by lane half
    const float* Sa = S + (h0 + m) * LDH;

    v8f c = {};
#pragma unroll
    for (int s0 = 0; s0 < NN; s0 += 4) {
        v2f a, bv;
        a.x  = Sa[s0 + 2 * kh];
        a.y  = Sa[s0 + 2 * kh + 1];
        bv.x = Q[(s0 + 2 * kh)     * LDH + h0 + m];
        bv.y = Q[(s0 + 2 * kh + 1) * LDH + h0 + m];
        // D = A(16x4) x B(4x16) + C, fp32 exact
        c = __builtin_amdgcn_wmma_f32_16x16x4_f32(false, a, false, bv,
                                                  (short)0, c, false, false);
    }

    // Diagonal of the 16x16 tile: VGPR r holds (M = r + 8*half, N = m).
    // M==N  =>  lanes 0..7 (r=m) and lanes 24..31 (r=m-8).
    const int r = m & 7;
    float dv = c[0];
    switch (r) {
        case 1: dv = c[1]; break;
        case 2: dv = c[2]; break;
        case 3: dv = c[3]; break;
        case 4: dv = c[4]; break;
        case 5: dv = c[5]; break;
        case 6: dv = c[6]; break;
        case 7: dv = c[7]; break;
        default: break;
    }
    if ((lane < 8) || (lane >= 24)) {
        const int h = h0 + m;
        const size_t off = (size_t)bd * NN + h;
        q_out[off] = s_arc[off] + dv;
    }
}

// ---------------------------------------------------------------------------
// Pipeline: (softmax -> contract) x3, then final softmax into d_out.
// Workspace: qs buffer + q buffer (each B*N*N floats), fully rewritten
// every call -> deterministic.
// ---------------------------------------------------------------------------
extern "C" void kernel_launch(void* const* d_in, const int* in_sizes, int n_in,
                              void* d_out, int out_size, void* d_ws, size_t ws_size,
                              hipStream_t stream) {
    const float*         s_arc = (const float*)d_in[0];
    const float*         s_sib = (const float*)d_in[1];
    const unsigned char* mask  = (const unsigned char*)d_in[2];
    float*               out   = (float*)d_out;

    float* qsbuf = (float*)d_ws;                 // [B,N,N] softmax probabilities
    float* qbuf  = qsbuf + (size_t)BB * NN * NN; // [B,N,N] updated scores

    const int rows = BB * NN;                    // 384
    const dim3 smGrid(rows / 4), smBlk(128);     // 4 wave32 rows per block
    const dim3 cGrid(rows),      cBlk(192);      // 6 wave32s, one h-tile each
    const size_t lds = (size_t)(2 * NN * LDH + NN) * sizeof(float); // ~74.9 KB

    // iter 1
    mfvi_softmax <<<smGrid, smBlk, 0,   stream>>>(s_arc, qsbuf, rows);
    mfvi_contract<<<cGrid,  cBlk,  lds, stream>>>(s_arc, s_sib, mask, qsbuf, qbuf);
    // iter 2
    mfvi_softmax <<<smGrid, smBlk, 0,   stream>>>(qbuf, qsbuf, rows);
    mfvi_contract<<<cGrid,  cBlk,  lds, stream>>>(s_arc, s_sib, mask, qsbuf, qbuf);
    // iter 3
    mfvi_softmax <<<smGrid, smBlk, 0,   stream>>>(qbuf, qsbuf, rows);
    mfvi_contract<<<cGrid,  cBlk,  lds, stream>>>(s_arc, s_sib, mask, qsbuf, qbuf);
    // final marginals
    mfvi_softmax <<<smGrid, smBlk, 0,   stream>>>(qbuf, out, rows);
}